// GPPTPrompt_49478023250330
// MI455X (gfx1250) — compile-verified
//
#include <hip/hip_runtime.h>

#define DHID 512
#define CN   16     // experts / router width
#define NC   10     // classes
#define NOUT 176    // 16 router cols + 160 expert*class cols

typedef float v2f __attribute__((ext_vector_type(2)));
typedef float v8f __attribute__((ext_vector_type(8)));

// ---------------------------------------------------------------------------
// Phase 0: self-loop init. hm = h, cnt = 1.0
// ---------------------------------------------------------------------------
__global__ void init_kernel(const float4* __restrict__ h, float4* __restrict__ hm,
                            float* __restrict__ cnt, int n4, int n) {
    int i = blockIdx.x * blockDim.x + threadIdx.x;
    if (i < n4) hm[i] = h[i];
    if (i < n)  cnt[i] = 1.0f;
}

// ---------------------------------------------------------------------------
// Phase 1a: in-degree counts (float, matches reference segment_sum of ones)
// ---------------------------------------------------------------------------
__global__ void count_kernel(const int* __restrict__ ei, float* __restrict__ cnt, int E) {
    int e = blockIdx.x * blockDim.x + threadIdx.x;
    if (e >= E) return;
    int dst = ei[E + e];
    __hip_atomic_fetch_add(&cnt[dst], 1.0f, __ATOMIC_RELAXED, __HIP_MEMORY_SCOPE_AGENT);
}

// ---------------------------------------------------------------------------
// Phase 1b: edge scatter-add. 64 consecutive threads = one edge; each thread
// moves 8 consecutive floats (two float4 loads from h[src], 8 f32 atomics to
// hm[dst]). h and hm are L2-resident (102 MB each vs 192 MB L2).
// ---------------------------------------------------------------------------
__global__ void scatter_kernel(const float* __restrict__ h, const int* __restrict__ ei,
                               float* __restrict__ hm, int E) {
    long long t = (long long)blockIdx.x * blockDim.x + threadIdx.x;
    int e = (int)(t >> 6);
    int c = (int)(t & 63);
    if (e >= E) return;
    int src = ei[e];
    int dst = ei[E + e];
    const float* hp = h  + (long long)src * DHID + c * 8;
    float*       op = hm + (long long)dst * DHID + c * 8;
    float4 a = *(const float4*)hp;
    float4 b = *(const float4*)(hp + 4);
    __hip_atomic_fetch_add(op + 0, a.x, __ATOMIC_RELAXED, __HIP_MEMORY_SCOPE_AGENT);
    __hip_atomic_fetch_add(op + 1, a.y, __ATOMIC_RELAXED, __HIP_MEMORY_SCOPE_AGENT);
    __hip_atomic_fetch_add(op + 2, a.z, __ATOMIC_RELAXED, __HIP_MEMORY_SCOPE_AGENT);
    __hip_atomic_fetch_add(op + 3, a.w, __ATOMIC_RELAXED, __HIP_MEMORY_SCOPE_AGENT);
    __hip_atomic_fetch_add(op + 4, b.x, __ATOMIC_RELAXED, __HIP_MEMORY_SCOPE_AGENT);
    __hip_atomic_fetch_add(op + 5, b.y, __ATOMIC_RELAXED, __HIP_MEMORY_SCOPE_AGENT);
    __hip_atomic_fetch_add(op + 6, b.z, __ATOMIC_RELAXED, __HIP_MEMORY_SCOPE_AGENT);
    __hip_atomic_fetch_add(op + 7, b.w, __ATOMIC_RELAXED, __HIP_MEMORY_SCOPE_AGENT);
}

// ---------------------------------------------------------------------------
// Phase 2: fused normalize + GEMM(176 cols via f32 WMMA) + argmax + select.
// One wave (32 lanes) per 16-node row tile. A-frag per ISA f32 16x4 layout,
// B-frag per-lane from the combined [176][512] weight rows (L2-resident).
// ---------------------------------------------------------------------------
__launch_bounds__(32)
__global__ void fused_kernel(const float* __restrict__ hm, const float* __restrict__ cnt,
                             const float* __restrict__ Ws, const float* __restrict__ Wt,
                             float* __restrict__ out) {
    __shared__ float hmLds[16 * DHID];   // 32 KB normalized A tile
    __shared__ float scLds[16 * NOUT];   // 11 KB score tile
    __shared__ float invLds[16];

    const int l = threadIdx.x;
    const long long node0 = (long long)blockIdx.x * 16;

    if (l < 16) invLds[l] = 1.0f / cnt[node0 + l];
    __syncthreads();

    // Cooperative load + normalize: 2048 float4 / 32 lanes = 64 each.
    #pragma unroll
    for (int i = 0; i < 64; ++i) {
        int flat = i * 32 + l;        // float4 index within tile
        int row  = flat >> 7;         // / 128 float4 per row
        int c4   = flat & 127;
        float4 v = *(const float4*)(hm + (node0 + row) * DHID + c4 * 4);
        float  s = invLds[row];
        float* d = &hmLds[row * DHID + c4 * 4];
        d[0] = v.x * s; d[1] = v.y * s; d[2] = v.z * s; d[3] = v.w * s;
    }
    __syncthreads();

    const int lm   = l & 15;
    const int koff = (l >= 16) ? 2 : 0;   // f32 A/B frag: lane half selects K pair

    // Per-lane weight row pointers: col = t*16 + lm over [W_struct ; W_task].
    const float* wp[11];
    #pragma unroll
    for (int t = 0; t < 11; ++t) {
        int col = t * 16 + lm;
        wp[t] = (col < CN) ? (Ws + col * DHID) : (Wt + (col - CN) * DHID);
    }

    v8f acc[11];
    #pragma unroll
    for (int t = 0; t < 11; ++t) acc[t] = (v8f){0.f,0.f,0.f,0.f,0.f,0.f,0.f,0.f};

    const float* arow = &hmLds[lm * DHID];
    for (int k4 = 0; k4 < DHID / 4; ++k4) {
        int kk = k4 * 4 + koff;
        v2f a;
        a.x = arow[kk];
        a.y = arow[kk + 1];
        #pragma unroll
        for (int t = 0; t < 11; ++t) {
            v2f b;
            b.x = wp[t][kk];
            b.y = wp[t][kk + 1];
            acc[t] = __builtin_amdgcn_wmma_f32_16x16x4_f32(
                false, a, false, b, (short)0, acc[t], false, false);
        }
    }

    // C layout: VGPR r -> row r (lanes 0-15) / row r+8 (lanes 16-31), col = lm.
    const int rbase = (l >= 16) ? 8 : 0;
    #pragma unroll
    for (int t = 0; t < 11; ++t) {
        #pragma unroll
        for (int r = 0; r < 8; ++r) {
            scLds[(rbase + r) * NOUT + t * 16 + lm] = acc[t][r];
        }
    }
    __syncthreads();

    if (l < 16) {
        const float* srow = &scLds[l * NOUT];
        int best = 0;
        float bv = srow[0];
        #pragma unroll
        for (int j = 1; j < CN; ++j) {      // strict > keeps first max (jnp.argmax)
            float v = srow[j];
            if (v > bv) { bv = v; best = j; }
        }
        const float* tv = srow + CN + best * NC;
        float* op = out + (node0 + l) * NC;
        #pragma unroll
        for (int c = 0; c < NC; ++c) op[c] = tv[c];
    }
}

// ---------------------------------------------------------------------------
extern "C" void kernel_launch(void* const* d_in, const int* in_sizes, int n_in,
                              void* d_out, int out_size, void* d_ws, size_t ws_size,
                              hipStream_t stream) {
    const float* h  = (const float*)d_in[0];
    const int*   ei = (const int*)d_in[1];
    const float* Ws = (const float*)d_in[2];   // [16, 512]
    const float* Wt = (const float*)d_in[3];   // [16, 10, 512] = [160, 512]
    float* out = (float*)d_out;

    const int n = in_sizes[0] / DHID;   // 50000
    const int E = in_sizes[1] / 2;      // 800000

    float* hm  = (float*)d_ws;                   // n*512 f32
    float* cnt = hm + (size_t)n * DHID;          // n f32

    const int n4 = n * (DHID / 4);
    init_kernel<<<(n4 + 255) / 256, 256, 0, stream>>>((const float4*)h, (float4*)hm, cnt, n4, n);
    count_kernel<<<(E + 255) / 256, 256, 0, stream>>>(ei, cnt, E);

    long long st = (long long)E * 64;
    scatter_kernel<<<(unsigned)((st + 255) / 256), 256, 0, stream>>>(h, ei, hm, E);

    fused_kernel<<<n / 16, 32, 0, stream>>>(hm, cnt, Ws, Wt, out);
}